// GlacierDynamicsCheckpointed_3659312136577
// MI455X (gfx1250) — compile-verified
//
#include <hip/hip_runtime.h>
#include <hip/hip_bf16.h>

// GlacierDynamicsCheckpointed for MI455X (gfx1250).
// Launch-latency-minimized schedule: kInit (state + initial SMB), then per
// step [kDiff: D + global max via per-step atomic slot] -> [kUpdate: flux
// update + snapshots + fused SMB refresh + last-block scalar update].
// State ping-pongs in d_ws (~14MB, resident in the 192MB L2). Halo tiles are
// staged global->LDS with the CDNA5 Tensor Data Mover (tensor_load_to_lds +
// s_wait_tensorcnt), D# descriptors per CDNA5 ISA ch.8 (2D, 4B elements).

#define GNY 768
#define GNX 768
#define GN2 (GNY * GNX)
#define NSTEPS 250
#define BLK_X 32
#define BLK_Y 8
#define NBLK_X 24   // 24*32 = 768
#define NBLK_Y 96   // 96*8  = 768
#define NBLOCKS (NBLK_X * NBLK_Y)

#ifndef USE_TDM
#if defined(__HIP_DEVICE_COMPILE__) && __has_builtin(__builtin_amdgcn_tensor_load_to_lds) && __has_builtin(__builtin_amdgcn_s_wait_tensorcnt)
#define USE_TDM 1
#else
#define USE_TDM 0
#endif
#endif

#if !USE_TDM && defined(__HIP_DEVICE_COMPILE__)
#warning "CDNA5 TDM builtins unavailable on device pass: using cooperative LDS fallback"
#endif

struct ScalState {
  float time;
  float t_last;
  int   idx;
  int   taken[6];
};

__device__ __forceinline__ int imin_(int a, int b) { return a < b ? a : b; }
__device__ __forceinline__ int imax_(int a, int b) { return a > b ? a : b; }

// FD * (RHO*G)^3, folded in double then cast (matches Python scalar fold)
__device__ __forceinline__ float sia_cd() {
  constexpr double rg = 910.0 * 9.81;
  constexpr double cd = 1.0e-16 * rg * rg * rg;
  return (float)cd;
}

// Degree-day SMB for one cell (update_smb + masked_smb from the reference).
__device__ __forceinline__ float smb_cell(float zsurf, float Tm, float Ts,
                                          float mf, float pr, float mk) {
  const float base = Tm - 6.5e-3f * zsurf;
  float meltsum = 0.0f, accsum = 0.0f;
#pragma unroll
  for (int m = 0; m < 12; m++) {
    const float T = base + Ts * sinf(6.283185307179586f * ((float)m + 0.5f) * (1.0f / 12.0f));
    meltsum += fmaxf(T, 0.0f);
    accsum += (T <= 0.0f) ? 1.0f : 0.0f;
  }
  const float s = (pr * (accsum * (1.0f / 12.0f)) -
                   mf * 365.0f * (meltsum * (1.0f / 12.0f))) * mk;
  return ((s < 0.0f) || (mk > 0.5f)) ? s : -10.0f;
}

// ---------------------------------------------------------------------------
// Tile staging: TDM (CDNA5) with cooperative-load fallback
// ---------------------------------------------------------------------------
__device__ __forceinline__ void coop_load_tile(float* lds, const float* origin,
                                               int tw, int th, int stride,
                                               int tid, int nthreads) {
  const int n = tw * th;
  for (int t = tid; t < n; t += nthreads) {
    int r = t / tw;
    int c = t - r * tw;
    lds[t] = origin[r * stride + c];
  }
}

#if USE_TDM
typedef unsigned int uint4t __attribute__((ext_vector_type(4)));
typedef int          int8t  __attribute__((ext_vector_type(8)));
typedef int          int4t  __attribute__((ext_vector_type(4)));

__device__ __forceinline__ unsigned lds_byte_off(const void* p) {
  // generic pointer to LDS: low 32 bits are the workgroup-relative LDS offset
  return (unsigned)(unsigned long long)p;
}

// Build a 2D D# (ISA 8.3/8.4) and issue TENSOR_LOAD_TO_LDS. All inputs are
// wave-uniform -> descriptor groups live in SGPRs as the encoding requires.
__device__ __forceinline__ void tdm_load_tile(unsigned ldsoff, const float* origin,
                                              int tile_w, int tile_h,
                                              int stride_elems) {
  unsigned long long ga = (unsigned long long)(const void*)origin;
  uint4t g0;
  g0[0] = 1u;                                                   // count=1, user D#
  g0[1] = ldsoff;                                               // lds_addr (bytes)
  g0[2] = (unsigned)(ga & 0xFFFFFFFFull);                       // global_addr[31:0]
  g0[3] = (unsigned)((ga >> 32) & 0x01FFFFFFull) | 0x80000000u; // addr[56:32] | type=2
  int8t g1;
  g1[0] = (int)(2u << 16);                                      // data_size=4B
  g1[1] = (int)(((unsigned)GNX & 0xFFFFu) << 16);               // tensor_dim0 lo16
  g1[2] = (int)(((unsigned)GNY & 0xFFFFu) << 16);               // tensor_dim1 lo16 | dim0 hi16(=0)
  g1[3] = (int)(((unsigned)tile_w & 0xFFFFu) << 16);            // tile_dim0 | dim1 hi16(=0)
  g1[4] = (int)((unsigned)tile_h & 0xFFFFu);                    // tile_dim1 (tile_dim2=0)
  g1[5] = stride_elems;                                         // tensor_dim0_stride lo32
  g1[6] = 0;
  g1[7] = 0;
  int4t gz = (int4t)0;                                          // groups 2/3 unused (2D)
#if defined(__clang_major__) && (__clang_major__ >= 23)
  int8t g5 = (int8t)0;
  __builtin_amdgcn_tensor_load_to_lds(g0, g1, gz, gz, g5, 0);
#else
  __builtin_amdgcn_tensor_load_to_lds(g0, g1, gz, gz, 0);
#endif
}
#endif // USE_TDM

// ---------------------------------------------------------------------------
// kInit: state buffers, zero outputs/slots, seed scalars, initial SMB (idx=0)
// ---------------------------------------------------------------------------
__global__ __launch_bounds__(256) void kInit(const float* __restrict__ Hini,
                                             const float* __restrict__ Ztopo,
                                             const float* __restrict__ precip,
                                             const float* __restrict__ Tml,
                                             const float* __restrict__ Tsp,
                                             const float* __restrict__ mfp,
                                             const float* __restrict__ mask,
                                             float* __restrict__ H0,
                                             float* __restrict__ Zs0,
                                             float* __restrict__ smb,
                                             float* __restrict__ out,
                                             unsigned int* __restrict__ slots,
                                             ScalState* sc) {
  int g = blockIdx.x * 256 + threadIdx.x;
  if (g < GN2) {
    const float h = Hini[g];
    const float zs = Ztopo[g] + h;
    H0[g]  = h;
    Zs0[g] = zs;
    smb[g] = smb_cell(zs, Tml[0], Tsp[0], mfp[0], precip[g], mask[g]);
#pragma unroll
    for (int k = 0; k < 7; k++) out[k * GN2 + g] = 0.0f;
  }
  if (g < 512) slots[g] = 0u;  // 256 max-slots + 256 counters
  if (g == 0) {
    sc->time = 1860.0f;
    sc->t_last = 1860.0f;
    sc->idx = 0;
#pragma unroll
    for (int k = 0; k < 6; k++) sc->taken[k] = 0;
  }
}

// ---------------------------------------------------------------------------
// kDiff: D on the corner grid + per-step global max (atomic slot)
// ---------------------------------------------------------------------------
__global__ __launch_bounds__(256) void kDiff(const float* __restrict__ H,
                                             const float* __restrict__ Zs,
                                             float* __restrict__ D,
                                             unsigned int* __restrict__ maxslot) {
  const int tx = threadIdx.x, ty = threadIdx.y;
  const int tid = tx + ty * BLK_X;
  const int c0 = blockIdx.x * BLK_X, r0 = blockIdx.y * BLK_Y;
  // 9x33 tile of Zs and H (corner (r,c) needs rows r..r+1, cols c..c+1),
  // origin clamped so every read is in-bounds.
  const int oc = imin_(c0, GNX - 33);
  const int orow = imin_(r0, GNY - 9);
  __shared__ float sZ[9 * 33];
  __shared__ float sH[9 * 33];
  __shared__ float wmax[BLK_Y];

#if USE_TDM
  if (ty == 0) {  // one DMA issue per block (wave 0)
    tdm_load_tile(lds_byte_off(sZ), Zs + orow * GNX + oc, 33, 9, GNX);
    tdm_load_tile(lds_byte_off(sH), H + orow * GNX + oc, 33, 9, GNX);
    __builtin_amdgcn_s_wait_tensorcnt(0);
  }
  __syncthreads();
#else
  coop_load_tile(sZ, Zs + orow * GNX + oc, 33, 9, GNX, tid, 256);
  coop_load_tile(sH, H + orow * GNX + oc, 33, 9, GNX, tid, 256);
  __syncthreads();
#endif

  const int r = r0 + ty, c = c0 + tx;
  float Dv = 0.0f;
  if (r < GNY - 1 && c < GNX - 1) {
    const int lr = r - orow, lc = c - oc;
    const float z00 = sZ[lr * 33 + lc];
    const float z01 = sZ[lr * 33 + lc + 1];
    const float z10 = sZ[(lr + 1) * 33 + lc];
    const float z11 = sZ[(lr + 1) * 33 + lc + 1];
    const float dsdx0 = (z01 - z00) * 0.01f;       // dSdx[r,c]
    const float dsdx1 = (z11 - z10) * 0.01f;       // dSdx[r+1,c]
    const float dsdy0 = (z10 - z00) * 0.01f;       // dSdy[r,c]
    const float dsdy1 = (z11 - z01) * 0.01f;       // dSdy[r,c+1]
    const float Sx = 0.5f * (dsdx1 + dsdx0);
    const float Sy = 0.5f * (dsdy1 + dsdy0);
    const float Havg = 0.25f * (sH[lr * 33 + lc] + sH[lr * 33 + lc + 1] +
                                sH[(lr + 1) * 33 + lc] + sH[(lr + 1) * 33 + lc + 1]);
    const float h2 = Havg * Havg;
    const float h5 = h2 * h2 * Havg;
    Dv = sia_cd() * h5 * (Sx * Sx + Sy * Sy);
    D[r * GNX + c] = Dv;
  }
  // wave32 max reduction, then block, then float-bits atomicMax (D >= 0)
  float m = Dv;
#pragma unroll
  for (int o = 16; o > 0; o >>= 1) m = fmaxf(m, __shfl_down(m, o, 32));
  if (tx == 0) wmax[ty] = m;
  __syncthreads();
  if (tid == 0) {
    float mm = wmax[0];
#pragma unroll
    for (int k = 1; k < BLK_Y; k++) mm = fmaxf(mm, wmax[k]);
    atomicMax(maxslot, __float_as_uint(mm));
  }
}

// ---------------------------------------------------------------------------
// kUpdate: dt from max(D), flux-divergence update, snapshots, fused SMB
// refresh (uses this thread's new Zs + incremented idx), optional final-H
// write, last-block scalar update.
// ---------------------------------------------------------------------------
__global__ __launch_bounds__(256) void kUpdate(const float* __restrict__ Hold,
                                               const float* __restrict__ Zsold,
                                               const float* __restrict__ Dg,
                                               const float* __restrict__ Ztopo,
                                               const float* __restrict__ precip,
                                               const float* __restrict__ Tml,
                                               const float* __restrict__ Tsp,
                                               const float* __restrict__ mfp,
                                               const float* __restrict__ mask,
                                               float* __restrict__ smb,
                                               float* __restrict__ Hnew,
                                               float* __restrict__ Zsnew,
                                               float* __restrict__ out,
                                               const unsigned int* __restrict__ maxslot,
                                               ScalState* sc,
                                               unsigned int* __restrict__ counter,
                                               int write_final) {
  const int tx = threadIdx.x, ty = threadIdx.y;
  const int tid = tx + ty * BLK_X;
  const int j0 = blockIdx.x * BLK_X, i0 = blockIdx.y * BLK_Y;
  // Zs halo tile 10x34, D 2x2-corner tile 9x33; origins clamped in-bounds.
  const int ocz = imin_(imax_(j0 - 1, 0), GNX - 34);
  const int orz = imin_(imax_(i0 - 1, 0), GNY - 10);
  const int ocd = imin_(imax_(j0 - 1, 0), (GNX - 1) - 33);
  const int ord = imin_(imax_(i0 - 1, 0), (GNY - 1) - 9);
  __shared__ float sZ[10 * 34];
  __shared__ float sD[9 * 33];

#if USE_TDM
  if (ty == 0) {
    tdm_load_tile(lds_byte_off(sZ), Zsold + orz * GNX + ocz, 34, 10, GNX);
    tdm_load_tile(lds_byte_off(sD), Dg + ord * GNX + ocd, 33, 9, GNX);
    __builtin_amdgcn_s_wait_tensorcnt(0);
  }
  __syncthreads();
#else
  coop_load_tile(sZ, Zsold + orz * GNX + ocz, 34, 10, GNX, tid, 256);
  coop_load_tile(sD, Dg + ord * GNX + ocd, 33, 9, GNX, tid, 256);
  __syncthreads();
#endif

  // scalar state (stable during this kernel; only the last block rewrites it)
  const float tcur = sc->time;
  const float tlast = sc->t_last;
  const int idx0 = sc->idx;
  int takenv[6];
#pragma unroll
  for (int k = 0; k < 6; k++) takenv[k] = sc->taken[k];

  const float maxD = __uint_as_float(*maxslot);
  float dt = fminf(2.0f, 4900.0f / (maxD + 1e-8f));  // 0.49*min(DX,DY)^2
  if (tcur >= 2060.0f) dt = 0.0f;                    // freeze at TTOT
  const float tnew = tcur + dt;
  const int do_upd = ((tnew - tlast) >= 10.0f) ? 1 : 0;  // identical on all threads

  const int i = i0 + ty, j = j0 + tx;
  const int g = i * GNX + j;
  const float hold = Hold[g];
  float hn;
  if (i >= 1 && i <= GNY - 2 && j >= 1 && j <= GNX - 2) {
    const int lr = i - orz, lc = j - ocz;
    const float zc = sZ[lr * 34 + lc];
    const float zw = sZ[lr * 34 + lc - 1];
    const float ze = sZ[lr * 34 + lc + 1];
    const float zn = sZ[(lr - 1) * 34 + lc];
    const float zs = sZ[(lr + 1) * 34 + lc];
    const float sxw = (zc - zw) * 0.01f;  // dSdx[i,j-1]
    const float sxe = (ze - zc) * 0.01f;  // dSdx[i,j]
    const float syn = (zc - zn) * 0.01f;  // dSdy[i-1,j]
    const float sys = (zs - zc) * 0.01f;  // dSdy[i,j]
    const int dr = i - ord, dc = j - ocd;
    const float Dnw = sD[(dr - 1) * 33 + dc - 1];
    const float Dne = sD[(dr - 1) * 33 + dc];
    const float Dsw = sD[dr * 33 + dc - 1];
    const float Dse = sD[dr * 33 + dc];
    const float qxe = -0.5f * (Dse + Dne) * sxe;
    const float qxw = -0.5f * (Dsw + Dnw) * sxw;
    const float qys = -0.5f * (Dse + Dsw) * sys;
    const float qyn = -0.5f * (Dne + Dnw) * syn;
    const float divq = (qxe - qxw) * 0.01f + (qys - qyn) * 0.01f;
    hn = hold + dt * (smb[g] - divq);
  } else {
    hn = hold;
  }
  hn = fmaxf(hn, 0.0f);
  const float zsn = Ztopo[g] + hn;
  Hnew[g] = hn;
  Zsnew[g] = zsn;
  if (write_final) out[6 * GN2 + g] = hn;

  // snapshots: first step whose post-step time crosses each snap year
  const float SNAPS[6] = {1880.0f, 1926.0f, 1957.0f, 1980.0f, 1999.0f, 2009.0f};
#pragma unroll
  for (int k = 0; k < 6; k++) {
    if (!takenv[k] && tnew >= SNAPS[k]) out[k * GN2 + g] = hn;
  }

  // fused SMB refresh: reference recomputes smb from the post-step Zs and the
  // incremented T_m index; each thread owns its cell, no cross-thread hazard.
  if (do_upd) {
    const int idx1 = imin_(imax_(idx0 + 1, 0), 31);
    smb[g] = smb_cell(zsn, Tml[idx1], Tsp[0], mfp[0], precip[g], mask[g]);
  }

  // last-block-done scalar update (per-step counter pre-zeroed by kInit)
  __threadfence();
  __syncthreads();
  if (tid == 0) {
    unsigned prev = atomicAdd(counter, 1u);
    if (prev == (unsigned)(NBLOCKS - 1)) {
#pragma unroll
      for (int k = 0; k < 6; k++)
        sc->taken[k] = (takenv[k] || (tnew >= SNAPS[k])) ? 1 : 0;
      sc->time = tnew;
      if (do_upd) {
        sc->t_last = tnew;
        sc->idx = idx0 + 1;
      }
      __threadfence();
    }
  }
}

// ---------------------------------------------------------------------------
// Host orchestration (graph-capture safe: only kernel launches on `stream`)
// ---------------------------------------------------------------------------
extern "C" void kernel_launch(void* const* d_in, const int* in_sizes, int n_in,
                              void* d_out, int out_size, void* d_ws, size_t ws_size,
                              hipStream_t stream) {
  (void)in_sizes; (void)n_in; (void)out_size; (void)ws_size;
  const float* precip = (const float*)d_in[0];
  const float* Tml    = (const float*)d_in[1];
  const float* Ts     = (const float*)d_in[2];
  // d_in[3]=P_daily, d_in[4]=T_daily: unused (parity with reference)
  const float* mf     = (const float*)d_in[5];
  const float* Ztopo  = (const float*)d_in[6];
  const float* Hini   = (const float*)d_in[7];
  const float* mask   = (const float*)d_in[8];
  float* out = (float*)d_out;

  // workspace layout: Zs0, Zs1, H0, H1, smb, D, slots(256), counters(256), scalars
  char* ws = (char*)d_ws;
  float* Zs[2] = {(float*)(ws + 0ull * GN2 * 4), (float*)(ws + 1ull * GN2 * 4)};
  float* H[2]  = {(float*)(ws + 2ull * GN2 * 4), (float*)(ws + 3ull * GN2 * 4)};
  float* smb   = (float*)(ws + 4ull * GN2 * 4);
  float* D     = (float*)(ws + 5ull * GN2 * 4);
  unsigned int* slots    = (unsigned int*)(ws + 6ull * GN2 * 4);
  unsigned int* counters = slots + 256;
  ScalState* sc = (ScalState*)(ws + 6ull * GN2 * 4 + 512 * 4);

  const dim3 blk(BLK_X, BLK_Y);
  const dim3 grd(NBLK_X, NBLK_Y);
  const int lin = (GN2 + 255) / 256;

  kInit<<<lin, 256, 0, stream>>>(Hini, Ztopo, precip, Tml, Ts, mf, mask,
                                 H[0], Zs[0], smb, out, slots, sc);

  for (int s = 0; s < NSTEPS; ++s) {
    const int p = s & 1, q = p ^ 1;
    kDiff<<<grd, blk, 0, stream>>>(H[p], Zs[p], D, slots + s);
    kUpdate<<<grd, blk, 0, stream>>>(H[p], Zs[p], D, Ztopo,
                                     precip, Tml, Ts, mf, mask, smb,
                                     H[q], Zs[q], out, slots + s, sc, counters + s,
                                     (s == NSTEPS - 1) ? 1 : 0);
  }
}